// PMSN_kernel_46145128628884
// MI455X (gfx1250) — compile-verified
//
#include <hip/hip_runtime.h>

typedef __attribute__((ext_vector_type(2))) float v2f;
typedef __attribute__((ext_vector_type(8))) float v8f;

#define N_DIM 4
#define L_FIX 8192   // reference L_LEN (compile-time per reference constants)

// sin/cos with explicit Cody-Waite mod-2pi reduction, then HW v_sin/v_cos.
// Valid for |x| up to ~1e6 with f32-reference-level accuracy (the reference
// itself forms the phase dtA_im*l in f32).
__device__ __forceinline__ void fast_sincos(float x, float* s, float* c) {
    const float INV_2PI  = 0.15915494309189535f;
    const float TWOPI_HI = 6.28318548202514648438f;   // fl(2*pi)
    const float TWOPI_LO = -1.7484556000744083e-7f;   // 2*pi - TWOPI_HI
    const float k = rintf(x * INV_2PI);
    float r = fmaf(-k, TWOPI_HI, x);
    r = fmaf(-k, TWOPI_LO, r);
    __sincosf(r, s, c);
}

// One block (256 thr = 8 waves) per h. Each wave computes 4 tiles of 16x16
// outputs (256 consecutive l each) via two chained V_WMMA_F32_16X16X4_F32.
__global__ __launch_bounds__(256) void pmsn_wmma_kernel(
    const float* __restrict__ log_dt,
    const float* __restrict__ log_A_real,
    const float* __restrict__ A_imag,
    const float* __restrict__ VinvB_real,
    const float* __restrict__ VinvB_imag,
    const float* __restrict__ CV_real,
    const float* __restrict__ CV_imag,
    float* __restrict__ out)
{
    const int h    = blockIdx.x;
    const int tid  = threadIdx.x;
    const int wave = tid >> 5;        // 0..7
    const int lane = tid & 31;
    const int half = lane >> 4;       // lane-half selects mode pair
    const int sub  = lane & 15;       // M (A rows), N (B cols / D cols)

    const float dt = __expf(log_dt[h]);

    // Per lane: modes n = half + 2*i  (i=0 -> WMMA#1 K0..3, i=1 -> WMMA#2 K4..7)
    float dtA_re[2], dtA_im[2], w_re[2], w_im[2];
    float q0[2], q1[2];               // B operand: Re(Abar^t), -Im(Abar^t)

#pragma unroll
    for (int i = 0; i < 2; ++i) {
        const int n   = half + 2 * i;
        const int idx = h * N_DIM + n;
        const float Ar = -__expf(log_A_real[idx]);
        const float Ai = A_imag[idx];
        const float dre = Ar * dt, dim = Ai * dt;
        dtA_re[i] = dre; dtA_im[i] = dim;

        // A_bar = exp(dt*A)
        float s, c;
        __sincosf(dim, &s, &c);
        const float er  = __expf(dre);
        const float abr = er * c, abi = er * s;

        // B_bar = (A_bar - 1) * B / A ;  w = C * B_bar
        const float Br = VinvB_real[idx], Bi = VinvB_imag[idx];
        const float t1r = (abr - 1.0f) * Br - abi * Bi;
        const float t1i = (abr - 1.0f) * Bi + abi * Br;
        const float inv_den = 1.0f / (Ar * Ar + Ai * Ai);
        const float bbr = (t1r * Ar + t1i * Ai) * inv_den;
        const float bbi = (t1i * Ar - t1r * Ai) * inv_den;
        const float Cr = CV_real[idx], Ci = CV_imag[idx];
        w_re[i] = Cr * bbr - Ci * bbi;
        w_im[i] = Cr * bbi + Ci * bbr;

        // Q column for this lane: t = sub (phase <= ~40 rad)
        const float t = (float)sub;
        float st, ct;
        fast_sincos(dim * t, &st, &ct);
        const float ert = __expf(dre * t);
        q0[i] =  ert * ct;
        q1[i] = -(ert * st);
    }

    float* outh = out + (size_t)h * L_FIX;

#pragma unroll
    for (int j = 0; j < 4; ++j) {
        const int tile   = wave * 4 + j;       // 0..31
        const int l0base = tile * 16;          // row-of-16 base index
        // A-operand row power: m = 16 * (l0base + M), M = sub
        const float m = (float)((l0base + sub) * 16);

        v8f acc = {};
#pragma unroll
        for (int i = 0; i < 2; ++i) {
            // P = w * Abar^m = w * exp(dtA * m); Cody-Waite + HW sincos
            float sm, cm;
            fast_sincos(dtA_im[i] * m, &sm, &cm);
            const float em = __expf(dtA_re[i] * m);
            const float kr = em * cm, ki = em * sm;
            v2f a, b;
            a.x = w_re[i] * kr - w_im[i] * ki;   // Re(P)  -> K = 2n
            a.y = w_re[i] * ki + w_im[i] * kr;   // Im(P)  -> K = 2n+1
            b.x = q0[i];                         // Re(Abar^t)
            b.y = q1[i];                         // -Im(Abar^t)
            acc = __builtin_amdgcn_wmma_f32_16x16x4_f32(
                false, a, false, b, (short)0, acc, false, false);
        }

        // D layout: reg v -> M = v + 8*half, N = sub. Fully coalesced b32 stores
        // (each half-wave writes one contiguous 64B segment per store).
        const int Mbase = half * 8;
#pragma unroll
        for (int v = 0; v < 8; ++v) {
            const int l = (l0base + v + Mbase) * 16 + sub;
            outh[l] = acc[v];
        }
    }
}

extern "C" void kernel_launch(void* const* d_in, const int* in_sizes, int n_in,
                              void* d_out, int out_size, void* d_ws, size_t ws_size,
                              hipStream_t stream) {
    (void)n_in; (void)out_size; (void)d_ws; (void)ws_size;
    const float* log_dt     = (const float*)d_in[0];
    const float* log_A_real = (const float*)d_in[1];
    const float* A_imag     = (const float*)d_in[2];
    const float* VinvB_real = (const float*)d_in[3];
    const float* VinvB_imag = (const float*)d_in[4];
    const float* CV_real    = (const float*)d_in[5];
    const float* CV_imag    = (const float*)d_in[6];
    float* out = (float*)d_out;

    const int H = in_sizes[0];   // 2048
    pmsn_wmma_kernel<<<H, 256, 0, stream>>>(
        log_dt, log_A_real, A_imag, VinvB_real, VinvB_imag, CV_real, CV_imag, out);
}